// ParallelMLPs_59777354826138
// MI455X (gfx1250) — compile-verified
//
#include <hip/hip_runtime.h>
#include <hip/hip_bf16.h>
#include <math.h>

// ---------------------------------------------------------------------------
// Fused ParallelMLPs for gfx1250 (MI455X, wave32, WMMA).
//   h = x @ hidden_w^T + hidden_b      -> FP32 WMMA 16x16x4, LDS-staged
//   a = act_block(h)                   -> uniform per 64-wide hidden tile
//   out[b,m,o] += sum_h a[b,h]*out_w[o,h]  -> LDS contraction + f32 atomics
//   out initialized to out_b broadcast by a tiny init kernel.
// ---------------------------------------------------------------------------

typedef __attribute__((ext_vector_type(2))) float v2f;
typedef __attribute__((ext_vector_type(8))) float v8f;

#define BATCH       1024
#define K_DIM       512
#define H_DIM       19200
#define O_DIM       10
#define N_MODELS    80
#define ACT_SPLIT   4800

#define TILE_M      128      // batch rows per workgroup
#define TILE_N      64       // hidden cols per workgroup (always 1 model, 1 act)
#define TILE_K      32       // K slab staged in LDS per iteration
#define LDA         36       // padded LDS stride (floats): 144B rows, 16B aligned
#define LDC         68       // padded stride for activated tile

#define NKT         (K_DIM / TILE_K)   // 16

union SharedMem {
    struct {
        float As[2][TILE_M * LDA];     // 2 x 18432 B
        float Bs[2][TILE_N * LDA];     // 2 x  9216 B
    } gemm;                            // 55296 B total
    struct {
        float a_s[TILE_M * LDC];       // 34816 B
        float w_s[O_DIM * TILE_N];     //  2560 B
    } proj;
};

__device__ __forceinline__ float activate(float v, int act) {
    switch (act) {
        case 0:  return fmaxf(v, 0.0f);                              // relu
        case 1:  return tanhf(v);                                    // tanh
        case 2:  return 1.0f / (1.0f + expf(-v));                    // sigmoid
        default: return 0.5f * v * (1.0f + erff(v * 0.7071067811865475f)); // gelu(exact)
    }
}

__global__ void __launch_bounds__(256)
parallel_mlps_fused(const float* __restrict__ x,
                    const float* __restrict__ hw,
                    const float* __restrict__ hb,
                    const float* __restrict__ ow,
                    const int*   __restrict__ nmid,
                    float*       __restrict__ out) {
    __shared__ SharedMem smem;

    const int tid   = threadIdx.x;
    const int wave  = tid >> 5;        // 0..7 (wave32)
    const int lane  = tid & 31;
    const int lmod  = lane & 15;       // M / N index within fragment
    const int lhalf = lane >> 4;       // selects K pair {0,1} vs {2,3}

    const int n0 = blockIdx.x * TILE_N;    // hidden tile start (multiple of 64)
    const int b0 = blockIdx.y * TILE_M;    // batch tile start

    const int act   = n0 / ACT_SPLIT;      // uniform activation for whole tile
    const int model = nmid[n0];            // uniform model id for whole tile

    v8f acc[4];
    #pragma unroll
    for (int nt = 0; nt < 4; ++nt) acc[nt] = (v8f){0,0,0,0,0,0,0,0};

    // -------- global -> register staging (float4 / global_load_b128) --------
    float4 xa[4];
    float4 wb[2];

    const float* xbase = x  + (size_t)b0 * K_DIM;
    const float* wbase = hw + (size_t)n0 * K_DIM;

    auto load_global = [&](int kt) {
        const float* xp = xbase + kt * TILE_K;
        #pragma unroll
        for (int i = 0; i < 4; ++i) {
            int idx = tid + 256 * i;            // 0..1023 -> 128 rows x 8 float4
            int row = idx >> 3, c4 = idx & 7;
            xa[i] = *(const float4*)(xp + (size_t)row * K_DIM + c4 * 4);
        }
        const float* wp = wbase + kt * TILE_K;
        #pragma unroll
        for (int i = 0; i < 2; ++i) {
            int idx = tid + 256 * i;            // 0..511 -> 64 rows x 8 float4
            int row = idx >> 3, c4 = idx & 7;
            wb[i] = *(const float4*)(wp + (size_t)row * K_DIM + c4 * 4);
        }
        // hint the next hidden_w slab toward L2 (global_prefetch_b8)
        if (kt + 2 < NKT)
            __builtin_prefetch(wbase + (kt + 2) * TILE_K + (size_t)(tid >> 3) * K_DIM, 0, 1);
    };

    auto store_lds = [&](int p) {
        #pragma unroll
        for (int i = 0; i < 4; ++i) {
            int idx = tid + 256 * i;
            int row = idx >> 3, c4 = idx & 7;
            *(float4*)&smem.gemm.As[p][row * LDA + c4 * 4] = xa[i];
        }
        #pragma unroll
        for (int i = 0; i < 2; ++i) {
            int idx = tid + 256 * i;
            int row = idx >> 3, c4 = idx & 7;
            *(float4*)&smem.gemm.Bs[p][row * LDA + c4 * 4] = wb[i];
        }
    };

    // ----------------------- WMMA main loop over K -------------------------
    auto compute = [&](int p) {
        const float* As = smem.gemm.As[p];
        const float* Bs = smem.gemm.Bs[p];
        #pragma unroll
        for (int kk = 0; kk < TILE_K; kk += 4) {
            const int k0 = kk + 2 * lhalf;     // even -> 8B-aligned ds_load_b64
            v2f a = *(const v2f*)&As[(wave * 16 + lmod) * LDA + k0];
            #pragma unroll
            for (int nt = 0; nt < 4; ++nt) {
                v2f b = *(const v2f*)&Bs[(nt * 16 + lmod) * LDA + k0];
                acc[nt] = __builtin_amdgcn_wmma_f32_16x16x4_f32(
                    false, a, false, b, (short)0, acc[nt], false, false);
            }
        }
    };

    load_global(0);
    int p = 0;
    for (int kt = 0; kt < NKT; ++kt) {
        store_lds(p);
        if (kt + 1 < NKT) load_global(kt + 1);
        __syncthreads();
        compute(p);
        p ^= 1;
    }
    __syncthreads();   // all GEMM LDS reads complete before union reuse

    // ------------- bias + activation, write activated tile to LDS ----------
    #pragma unroll
    for (int nt = 0; nt < 4; ++nt) {
        const int ncol = n0 + nt * 16 + lmod;
        const float bias = hb[ncol];
        #pragma unroll
        for (int r = 0; r < 8; ++r) {
            float v = acc[nt][r] + bias;
            v = activate(v, act);
            const int bl = wave * 16 + r + 8 * lhalf;   // local batch row
            smem.proj.a_s[bl * LDC + nt * 16 + lmod] = v;
        }
    }
    // stage out_w[0..9][n0..n0+63] (640 floats) with 160 threads
    if (tid < 160) {
        int o = tid >> 4, j = tid & 15;
        *(float4*)&smem.proj.w_s[o * TILE_N + j * 4] =
            *(const float4*)(ow + (size_t)o * H_DIM + n0 + j * 4);
    }
    __syncthreads();

    // ------------- out projection: 128 b x 10 o dots of length 64 ----------
    #pragma unroll
    for (int i = 0; i < 5; ++i) {
        const int pidx = tid + 256 * i;        // 0..1279
        const int b = pidx / O_DIM;
        const int o = pidx % O_DIM;
        const float* ap = &smem.proj.a_s[b * LDC];
        const float* wp = &smem.proj.w_s[o * TILE_N];
        float s = 0.0f;
        #pragma unroll
        for (int n = 0; n < TILE_N; ++n) s = fmaf(ap[n], wp[n], s);
        atomicAdd(&out[(size_t)(b0 + b) * (N_MODELS * O_DIM) + model * O_DIM + o], s);
    }
}

// out[b][m][o] = out_b[m][o]  (broadcast bias init; atomics accumulate on top)
__global__ void __launch_bounds__(256)
init_out_kernel(const float* __restrict__ ob, float* __restrict__ out) {
    const int idx = blockIdx.x * 256 + threadIdx.x;   // 819200 total
    out[idx] = ob[idx % (N_MODELS * O_DIM)];
}

extern "C" void kernel_launch(void* const* d_in, const int* in_sizes, int n_in,
                              void* d_out, int out_size, void* d_ws, size_t ws_size,
                              hipStream_t stream) {
    const float* x    = (const float*)d_in[0];   // [1024, 512]
    const float* hw   = (const float*)d_in[1];   // [19200, 512]
    const float* hb   = (const float*)d_in[2];   // [19200]
    const float* ow   = (const float*)d_in[3];   // [10, 19200]
    const float* ob   = (const float*)d_in[4];   // [80, 10]
    const int*   nmid = (const int*)d_in[5];     // [19200]
    float* out = (float*)d_out;                  // [1024, 80, 10]

    init_out_kernel<<<(BATCH * N_MODELS * O_DIM) / 256, 256, 0, stream>>>(ob, out);

    dim3 grid(H_DIM / TILE_N, BATCH / TILE_M);   // (300, 8)
    parallel_mlps_fused<<<grid, 256, 0, stream>>>(x, hw, hb, ow, nmid, out);
}